// YOLOv1_82635170775214
// MI455X (gfx1250) — compile-verified
//
#include <hip/hip_runtime.h>

#define NCLS 20
#define GRID7 7
#define MCELLS 49
#define FEAT 30
#define IMG_F (MCELLS * FEAT)   // 1470 floats per image
#define IOU_THR 0.7f
#define SCORE_THR 0.1f
#define IMGS_PER_BLOCK 8
#define PAD 52
#define SCR_PER_IMG (6 * PAD)   // 312 floats NMS scratch per image

// Output offsets (floats), outputs concatenated in reference return order.
#define O_LOGITS 0ull
#define O_YBB    16056320ull   // 16384*49*20
#define O_BOXES  20070400ull
#define O_SCORE  23281664ull
#define O_LABEL  24084480ull
#define O_KEEP   24887296ull
#define O_RES    25690112ull

typedef unsigned int u32x4 __attribute__((ext_vector_type(4)));
typedef int          i32x4 __attribute__((ext_vector_type(4)));
typedef int          i32x8 __attribute__((ext_vector_type(8)));

__device__ __forceinline__ unsigned rfl(unsigned v) {
  return (unsigned)__builtin_amdgcn_readfirstlane((int)v);
}

__global__ __launch_bounds__(256) void yolo_post_kernel(const float* __restrict__ x,
                                                        float* __restrict__ out,
                                                        int batch) {
  // Layout: 8 staged images contiguous (TDM 2-D tile target), then per-image scratch.
  __shared__ __align__(16) float smem[IMGS_PER_BLOCK * (IMG_F + SCR_PER_IMG)];
  const int lane = threadIdx.x & 31;
  const int wid  = threadIdx.x >> 5;
  const int img0 = blockIdx.x * IMGS_PER_BLOCK;
  const int rows = min(IMGS_PER_BLOCK, batch - img0);
  const int img  = img0 + wid;

  // ---- Phase 0: wave 0 issues ONE 2-D TDM descriptor staging all `rows` images ----
  if (wid == 0) {
    unsigned long long ga = (unsigned long long)(const void*)(x + (size_t)img0 * IMG_F);
    unsigned lds_off = (unsigned)(size_t)(void*)&smem[0];
    u32x4 g0;
    g0.x = 1u;                                                   // count=1, user D#
    g0.y = rfl(lds_off);                                         // lds_addr (bytes)
    g0.z = rfl((unsigned)(ga & 0xffffffffull));                  // global_addr lo
    g0.w = rfl((unsigned)((ga >> 32) & 0x1ffffffull) | (2u << 30));  // addr hi | type=2
    i32x8 g1;
    g1[0] = 0x20000;                       // data_size=2 (4 bytes)
    g1[1] = (IMG_F & 0xffff) << 16;        // tensor_dim0[15:0] = 1470
    g1[2] = rfl((unsigned)rows) << 16;     // tensor_dim1 = rows
    g1[3] = (IMG_F & 0xffff) << 16;        // tile_dim0 = 1470
    g1[4] = rfl((unsigned)rows);           // tile_dim1 = rows, tile_dim2 = 0
    g1[5] = IMG_F;                         // tensor_dim0_stride = 1470
    g1[6] = 0;
    g1[7] = 0;
    i32x4 z4 = {0, 0, 0, 0};
#if __clang_major__ >= 23
    i32x8 z8 = {0, 0, 0, 0, 0, 0, 0, 0};
    __builtin_amdgcn_tensor_load_to_lds(g0, g1, z4, z4, z8, 0);
#else
    __builtin_amdgcn_tensor_load_to_lds(g0, g1, z4, z4, 0);
#endif
    __builtin_amdgcn_s_wait_tensorcnt(0);
  }
  __syncthreads();  // publish TDM-written LDS to all waves
  if (wid >= rows) return;  // terminated waves still satisfy later barriers

  float* st    = &smem[wid * IMG_F];                               // staged image
  float* scr   = &smem[IMGS_PER_BLOCK * IMG_F + wid * SCR_PER_IMG];
  float* sA    = scr;            // adjusted scores (orig order)
  float* ssort = sA + PAD;       // sorted adjusted scores
  float* bsx0  = ssort + PAD;    // sorted offset boxes
  float* bsy0  = bsx0 + PAD;
  float* bsx1  = bsy0 + PAD;
  float* bsy1  = bsx1 + PAD;

  // ---- Phase 1: per-cell decode; lane owns cells {lane, lane+32} ----
  const size_t cellbase = (size_t)img * MCELLS;
  const int nc = (lane < MCELLS - 32) ? 2 : 1;

  float px0[2], py0[2], px1[2], py1[2], psc[2];
  int   plab[2], cellc[2], rnk[2];
  float maxc = 0.0f;

#pragma unroll 2
  for (int ci = 0; ci < nc; ++ci) {
    const int c = lane + ci * 32;
    cellc[ci] = c;
    const float* f = st + c * FEAT;
    // argmax(softmax(logits)) == argmax(logits); first-max tie-break
    float bv = f[0]; int lab = 0;
#pragma unroll
    for (int k = 1; k < NCLS; ++k) { float v = f[k]; if (v > bv) { bv = v; lab = k; } }
    // best of 2 boxes by confidence (argmax picks first on tie)
    const float c0 = f[NCLS + 4], c1 = f[NCLS + 9];
    const int best = (c1 > c0) ? 1 : 0;
    const float* bb = f + NCLS + best * 5;
    const float bx = bb[0], by = bb[1], bw = bb[2], bh = bb[3], conf = bb[4];
    const int gx = c % GRID7, gy = c / GRID7;
    const float cx = (bx + (float)gx) * (1.0f / (float)GRID7);
    const float cy = (by + (float)gy) * (1.0f / (float)GRID7);
    const float hw = bw * 0.5f, hh = bh * 0.5f;
    const float X0 = fminf(fmaxf((cx - hw) * 448.0f, 0.0f), 448.0f);
    const float Y0 = fminf(fmaxf((cy - hh) * 448.0f, 0.0f), 448.0f);
    const float X1 = fminf(fmaxf((cx + hw) * 448.0f, 0.0f), 448.0f);
    const float Y1 = fminf(fmaxf((cy + hh) * 448.0f, 0.0f), 448.0f);
    maxc = fmaxf(maxc, fmaxf(fmaxf(X0, Y0), fmaxf(X1, Y1)));

    const size_t cid = cellbase + c;
    float4* lo = reinterpret_cast<float4*>(out + O_LOGITS + cid * 20);  // 80B aligned
#pragma unroll
    for (int q = 0; q < 5; ++q)
      lo[q] = make_float4(f[4 * q], f[4 * q + 1], f[4 * q + 2], f[4 * q + 3]);
    float* yb = out + O_YBB + cid * 5;
    yb[0] = bx; yb[1] = by; yb[2] = bw; yb[3] = bh; yb[4] = conf;
    *reinterpret_cast<float4*>(out + O_BOXES + cid * 4) = make_float4(X0, Y0, X1, Y1);
    out[O_SCORE + cid] = conf;
    out[O_LABEL + cid] = (float)lab;
    const bool valid = conf > SCORE_THR;
    out[O_RES + cid] = valid ? 1.0f : 0.0f;

    px0[ci] = X0; py0[ci] = Y0; px1[ci] = X1; py1[ci] = Y1;
    psc[ci] = valid ? conf : -__builtin_inff();
    plab[ci] = lab;
  }

  // wave32 max-reduce of max box coordinate
#pragma unroll
  for (int o = 16; o > 0; o >>= 1) maxc = fmaxf(maxc, __shfl_xor(maxc, o, 32));
  const float offscale = maxc + 1.0f;

  float ax0[2], ay0[2], ax1[2], ay1[2];
#pragma unroll 2
  for (int ci = 0; ci < nc; ++ci) {
    const float offv = (float)plab[ci] * offscale;
    ax0[ci] = px0[ci] + offv; ay0[ci] = py0[ci] + offv;
    ax1[ci] = px1[ci] + offv; ay1[ci] = py1[ci] + offv;
    sA[cellc[ci]] = psc[ci];
  }
  __syncthreads();

  // ---- Phase 2: stable descending rank, scatter into sorted arrays ----
#pragma unroll 2
  for (int ci = 0; ci < nc; ++ci) {
    const int c = cellc[ci];
    const float si = psc[ci];
    int r = 0;
    for (int j = 0; j < MCELLS; ++j) {
      const float sj = sA[j];
      r += ((sj > si) || ((sj == si) && (j < c))) ? 1 : 0;
    }
    rnk[ci] = r;
    ssort[r] = si;
    bsx0[r] = ax0[ci]; bsy0[r] = ay0[ci]; bsx1[r] = ax1[ci]; bsy1[r] = ay1[ci];
  }
  __syncthreads();

  // ---- Phase 3: NMS scan; suppressed/keep sets as wave-uniform ballot masks ----
  float qx0[2] = {0, 0}, qy0[2] = {0, 0}, qx1[2] = {0, 0}, qy1[2] = {0, 0}, qa[2] = {0, 0};
  const bool qv1 = (lane < MCELLS - 32);
  {
    int q = lane;
    qx0[0] = bsx0[q]; qy0[0] = bsy0[q]; qx1[0] = bsx1[q]; qy1[0] = bsy1[q];
    qa[0] = fmaxf(qx1[0] - qx0[0], 0.0f) * fmaxf(qy1[0] - qy0[0], 0.0f);
    if (qv1) {
      q = lane + 32;
      qx0[1] = bsx0[q]; qy0[1] = bsy0[q]; qx1[1] = bsx1[q]; qy1[1] = bsy1[q];
      qa[1] = fmaxf(qx1[1] - qx0[1], 0.0f) * fmaxf(qy1[1] - qy0[1], 0.0f);
    }
  }
  unsigned sup0 = 0, sup1 = 0, keep0 = 0, keep1 = 0;
  for (int i = 0; i < MCELLS; ++i) {
    const bool sup_i = (i < 32) ? ((sup0 >> i) & 1u) : ((sup1 >> (i - 32)) & 1u);
    const bool v_i = ssort[i] > -1e30f;
    const bool keep_i = v_i && !sup_i;
    if (i < 32) keep0 |= ((unsigned)keep_i) << i;
    else        keep1 |= ((unsigned)keep_i) << (i - 32);
    const float ix0 = bsx0[i], iy0 = bsy0[i], ix1 = bsx1[i], iy1 = bsy1[i];
    const float ia = fmaxf(ix1 - ix0, 0.0f) * fmaxf(iy1 - iy0, 0.0f);
    bool s0, s1;
    {
      const float w = fmaxf(fminf(ix1, qx1[0]) - fmaxf(ix0, qx0[0]), 0.0f);
      const float h = fmaxf(fminf(iy1, qy1[0]) - fmaxf(iy0, qy0[0]), 0.0f);
      const float inter = w * h;
      const float uni = fmaxf(ia + qa[0] - inter, 1e-9f);
      s0 = keep_i && (lane > i) && (inter > IOU_THR * uni);
    }
    {
      const float w = fmaxf(fminf(ix1, qx1[1]) - fmaxf(ix0, qx0[1]), 0.0f);
      const float h = fmaxf(fminf(iy1, qy1[1]) - fmaxf(iy0, qy0[1]), 0.0f);
      const float inter = w * h;
      const float uni = fmaxf(ia + qa[1] - inter, 1e-9f);
      s1 = qv1 && keep_i && ((lane + 32) > i) && (inter > IOU_THR * uni);
    }
    sup0 |= (unsigned)(__ballot(s0) & 0xffffffffull);
    sup1 |= (unsigned)(__ballot(s1) & 0xffffffffull);
  }

  // ---- Phase 4: scatter keep mask back to original order ----
#pragma unroll 2
  for (int ci = 0; ci < nc; ++ci) {
    const int r = rnk[ci];
    const bool kept = (r < 32) ? ((keep0 >> r) & 1u) : ((keep1 >> (r - 32)) & 1u);
    out[O_KEEP + cellbase + cellc[ci]] = kept ? 1.0f : 0.0f;
  }
}

extern "C" void kernel_launch(void* const* d_in, const int* in_sizes, int n_in,
                              void* d_out, int out_size, void* d_ws, size_t ws_size,
                              hipStream_t stream) {
  const float* x = (const float*)d_in[0];
  float* out = (float*)d_out;
  const int batch = in_sizes[0] / IMG_F;  // 16384
  const int blocks = (batch + IMGS_PER_BLOCK - 1) / IMGS_PER_BLOCK;
  yolo_post_kernel<<<blocks, 256, 0, stream>>>(x, out, batch);
}